// Net_39831526703826
// MI455X (gfx1250) — compile-verified
//
#include <hip/hip_runtime.h>
#include <hip/hip_bf16.h>
#include <math.h>

// ============================================================================
// PointNet++ (SSG) forward for MI455X / gfx1250, fp32 end-to-end.
// GEMMs use V_WMMA_F32_16X16X4_F32 with compile-time strides and 16x(16*NT)
// per-wave tiles (A-load reuse across NT WMMAs). BN folded into next layer's
// weights. FPS done in one workgroup with min_d in LDS. Scratch need ~103MB.
// ============================================================================

typedef float v2f __attribute__((ext_vector_type(2)));
typedef float v8f __attribute__((ext_vector_type(8)));

#define EPS_BN 1e-5f

// ---------------------------------------------------------------------------
__global__ void zero_int_k(int* p) { *p = 0; }

// ---------------------------------------------------------------------------
__global__ void permute_k(const float* __restrict__ x, const float* __restrict__ pos,
                          const int* __restrict__ perm,
                          float* __restrict__ xp, float* __restrict__ pp, int N) {
  int i = blockIdx.x * blockDim.x + threadIdx.x;
  if (i >= N) return;
  int s = perm[i];
#pragma unroll
  for (int f = 0; f < 8; f++) xp[i * 8 + f] = x[s * 8 + f];
  pp[i * 3 + 0] = pos[s * 3 + 0];
  pp[i * 3 + 1] = pos[s * 3 + 1];
  pp[i * 3 + 2] = pos[s * 3 + 2];
}

// ---------------------------------------------------------------------------
// Farthest point sampling, single workgroup (1024 threads = 32 waves).
// min_d kept in LDS; per-step argmax via wave shfl + LDS tree.
__global__ void __launch_bounds__(1024) fps_k(const float* __restrict__ pos, int N,
                                              int nS, float* __restrict__ qout) {
  __shared__ float mind[8192];
  __shared__ float wv[32];
  __shared__ int   wi[32];
  __shared__ int   curS;
  int t = threadIdx.x, lane = t & 31, wid = t >> 5;
  for (int p = t; p < N; p += 1024) mind[p] = 3.4e38f;
  int cur = 0;
  __syncthreads();
  for (int s = 0; s < nS; s++) {
    float cx = pos[cur * 3], cy = pos[cur * 3 + 1], cz = pos[cur * 3 + 2];
    if (t == 0) { qout[s * 3] = cx; qout[s * 3 + 1] = cy; qout[s * 3 + 2] = cz; }
    float bv = -1.0f; int bi = 0;
    for (int p = t; p < N; p += 1024) {
      float dx = pos[p * 3] - cx, dy = pos[p * 3 + 1] - cy, dz = pos[p * 3 + 2] - cz;
      float d = dx * dx + dy * dy + dz * dz;
      float m = fminf(mind[p], d);
      mind[p] = m;
      if (m > bv) { bv = m; bi = p; }          // p increasing -> first max kept
    }
    for (int off = 16; off > 0; off >>= 1) {   // wave argmax (prefer lower idx)
      float ov = __shfl_down(bv, off, 32);
      int   oi = __shfl_down(bi, off, 32);
      if (ov > bv || (ov == bv && oi < bi)) { bv = ov; bi = oi; }
    }
    if (lane == 0) { wv[wid] = bv; wi[wid] = bi; }
    __syncthreads();
    if (wid == 0) {
      float v2 = wv[lane]; int i2 = wi[lane];
      for (int off = 16; off > 0; off >>= 1) {
        float ov = __shfl_down(v2, off, 32);
        int   oi = __shfl_down(i2, off, 32);
        if (ov > v2 || (ov == v2 && oi < i2)) { v2 = ov; i2 = oi; }
      }
      if (lane == 0) curS = i2;
    }
    __syncthreads();
    cur = curS;
  }
}

// ---------------------------------------------------------------------------
// Ball query: first 64 in-radius sources by index; -1 sentinel; count valid.
__global__ void ball_query_k(const float* __restrict__ ps, int Ns,
                             const float* __restrict__ pq, int Q, float r2,
                             int* __restrict__ nbr, int* __restrict__ cnt) {
  int q = blockIdx.x * blockDim.x + threadIdx.x;
  if (q >= Q) return;
  float qx = pq[q * 3], qy = pq[q * 3 + 1], qz = pq[q * 3 + 2];
  int c = 0;
  for (int j = 0; j < Ns; j++) {
    float dx = ps[j * 3] - qx, dy = ps[j * 3 + 1] - qy, dz = ps[j * 3 + 2] - qz;
    if (dx * dx + dy * dy + dz * dz <= r2) {
      nbr[q * 64 + c] = j;
      if (++c == 64) break;
    }
  }
  for (int k = c; k < 64; k++) nbr[q * 64 + k] = -1;
  atomicAdd(cnt, c);
}

// ---------------------------------------------------------------------------
// Gather kernels: build [rows x Kpad] A-matrices (zero-padded K columns).
__global__ void gather_sa1_k(const float* __restrict__ xp, const float* __restrict__ posp,
                             const float* __restrict__ q, const int* __restrict__ nbr,
                             float* __restrict__ H0) {
  int r = blockIdx.x * blockDim.x + threadIdx.x;
  if (r >= 4096 * 64) return;
  int qi = r >> 6, j = nbr[r];
  float* o = H0 + (size_t)r * 12;
  if (j >= 0) {
#pragma unroll
    for (int f = 0; f < 8; f++) o[f] = xp[j * 8 + f];
    o[8]  = posp[j * 3 + 0] - q[qi * 3 + 0];
    o[9]  = posp[j * 3 + 1] - q[qi * 3 + 1];
    o[10] = posp[j * 3 + 2] - q[qi * 3 + 2];
    o[11] = 0.0f;
  } else {
#pragma unroll
    for (int f = 0; f < 12; f++) o[f] = 0.0f;
  }
}

__global__ void gather_sa2_k(const float* __restrict__ x1, const float* __restrict__ pos1,
                             const float* __restrict__ q, const int* __restrict__ nbr,
                             float* __restrict__ H0) {
  int r = blockIdx.x * blockDim.x + threadIdx.x;
  if (r >= 512 * 64) return;
  int qi = r >> 6, j = nbr[r];
  float* o = H0 + (size_t)r * 68;
  if (j >= 0) {
    for (int f = 0; f < 64; f++) o[f] = x1[j * 64 + f];
    o[64] = pos1[j * 3 + 0] - q[qi * 3 + 0];
    o[65] = pos1[j * 3 + 1] - q[qi * 3 + 1];
    o[66] = pos1[j * 3 + 2] - q[qi * 3 + 2];
    o[67] = 0.0f;
  } else {
    for (int f = 0; f < 68; f++) o[f] = 0.0f;
  }
}

__global__ void concat_sa3_k(const float* __restrict__ x2, const float* __restrict__ pos2,
                             float* __restrict__ H0) {
  int i = blockIdx.x * blockDim.x + threadIdx.x;
  if (i >= 512 * 132) return;
  int r = i / 132, c = i - r * 132;
  float v;
  if (c < 128)      v = x2[r * 128 + c];
  else if (c < 131) v = pos2[r * 3 + (c - 128)];
  else              v = 0.0f;
  H0[i] = v;
}

// ---------------------------------------------------------------------------
// Fold previous layer's BN (s,t) into this layer's weights:
//   W'[k,n] = s[k]*W[k,n], b'[n] = b[n] + sum_k t[k]*W[k,n]; zero K-padding.
// Also zeroes the stats accumulators for the upcoming GEMM. blockDim = N.
__global__ void prep_wb_k(const float* __restrict__ W, const float* __restrict__ b,
                          const float* __restrict__ s, const float* __restrict__ t,
                          float* __restrict__ Wp, float* __restrict__ bp,
                          int K, int Kpad, int N,
                          float* __restrict__ gsum, float* __restrict__ gsq) {
  int n = threadIdx.x;
  float acc = b[n];
  for (int k = 0; k < Kpad; k++) {
    float w = 0.0f;
    if (k < K) {
      w = W[k * N + n];
      if (t) acc += t[k] * w;
      if (s) w *= s[k];
    }
    Wp[k * N + n] = w;
  }
  bp[n] = acc;
  gsum[n] = 0.0f;
  gsq[n]  = 0.0f;
}

// ---------------------------------------------------------------------------
// BN finalize: per-column scale/shift from masked sums. blockDim = N.
__global__ void fin_stats_k(const float* __restrict__ gsum, const float* __restrict__ gsq,
                            const int* __restrict__ cntPtr, int cntConst,
                            const float* __restrict__ gamma, const float* __restrict__ beta,
                            float* __restrict__ sv, float* __restrict__ tv) {
  int n = threadIdx.x;
  float cnt = cntPtr ? fmaxf((float)(*cntPtr), 1.0f) : (float)cntConst;
  float mean = gsum[n] / cnt;
  float var  = fmaxf(gsq[n] / cnt - mean * mean, 0.0f);
  float s = gamma[n] / sqrtf(var + EPS_BN);
  sv[n] = s;
  tv[n] = beta[n] - mean * s;
}

// ---------------------------------------------------------------------------
// GEMM: H = relu(A @ W' + b'), accumulate masked per-column sum/sumsq.
// One wave computes a 16 x (16*NT) output tile; each A load (v2f) feeds NT
// V_WMMA_F32_16X16X4_F32 ops. KPAD/N compile-time -> immediate-offset loads.
// A layout (ISA 16x4 f32): lane = M (0..15), half-wave selects K pair.
// C/D layout: element (vgpr v, lane L) = D[M = v + 8*(L>>4)][N = L&15].
template <int KPAD, int N, int NT>
__global__ void __launch_bounds__(32) mlp_gemm_t(
    const float* __restrict__ A,
    const float* __restrict__ Wp, const float* __restrict__ bp,
    float* __restrict__ H,
    const int* __restrict__ validFlat,
    float* __restrict__ gsum, float* __restrict__ gsq) {
  int lane = threadIdx.x;
  int half = lane >> 4;
  int l15  = lane & 15;
  int m0 = blockIdx.x * 16;
  int n0 = blockIdx.y * (16 * NT);

  const v8f vzero = {0.f, 0.f, 0.f, 0.f, 0.f, 0.f, 0.f, 0.f};
  v8f acc[NT];
#pragma unroll
  for (int i = 0; i < NT; i++) acc[i] = vzero;

  const float* Arow  = A  + (size_t)(m0 + l15) * KPAD + 2 * half;
  const float* Bbase = Wp + (size_t)(2 * half) * N + n0 + l15;

  for (int k = 0; k < KPAD; k += 4) {
    v2f a = *(const v2f*)(Arow + k);                  // A[m][k+2h..k+2h+1]
#pragma unroll
    for (int nt = 0; nt < NT; nt++) {
      v2f b;
      b.x = Bbase[(size_t)k * N       + nt * 16];     // B[k+2h  ][n]
      b.y = Bbase[(size_t)(k + 1) * N + nt * 16];     // B[k+2h+1][n]
      acc[nt] = __builtin_amdgcn_wmma_f32_16x16x4_f32(false, a, false, b,
                                                      (short)0, acc[nt],
                                                      false, false);
    }
  }

#pragma unroll
  for (int nt = 0; nt < NT; nt++) {
    int col = n0 + nt * 16 + l15;
    float bias = bp[col];
    float sum = 0.0f, sq = 0.0f;
#pragma unroll
    for (int v = 0; v < 8; v++) {
      int row = m0 + v + 8 * half;
      float h = fmaxf(acc[nt][v] + bias, 0.0f);
      H[(size_t)row * N + col] = h;
      bool ok = validFlat ? (validFlat[row] >= 0) : true;
      if (ok) { sum += h; sq += h * h; }
    }
    // lanes L and L^16 hold the same output column -> fold halves, then atomics
    sum += __shfl_xor(sum, 16, 32);
    sq  += __shfl_xor(sq, 16, 32);
    if (half == 0) {
      atomicAdd(&gsum[col], sum);
      atomicAdd(&gsq[col], sq);
    }
  }
}

// ---------------------------------------------------------------------------
// Masked max over K=64 neighbors, applying final-layer BN (s,t) elementwise.
__global__ void pc_max_k(const float* __restrict__ H, int C,
                         const int* __restrict__ nbr,
                         const float* __restrict__ s, const float* __restrict__ t,
                         float* __restrict__ X) {
  int q = blockIdx.x, c = threadIdx.x;      // blockDim = C
  float sc = s[c], tc = t[c], m = -3.4e38f;
  for (int k = 0; k < 64; k++) {
    int r = (q << 6) + k;
    if (nbr[r] >= 0) m = fmaxf(m, H[(size_t)r * C + c] * sc + tc);
  }
  X[q * C + c] = m;
}

__global__ void colmax_k(const float* __restrict__ H, int rows, int C,
                         const float* __restrict__ s, const float* __restrict__ t,
                         float* __restrict__ g) {
  int c = blockIdx.x * blockDim.x + threadIdx.x;
  if (c >= C) return;
  float sc = s[c], tc = t[c], m = -3.4e38f;
  for (int r = 0; r < rows; r++) m = fmaxf(m, H[(size_t)r * C + c] * sc + tc);
  g[c] = m;
}

// ---------------------------------------------------------------------------
// Head: feat = [g(512), y[1]] -> 256 -> 128 -> 1. Tiny; one block.
__global__ void __launch_bounds__(256) head_k(
    const float* __restrict__ g, const float* __restrict__ y,
    const float* __restrict__ W1, const float* __restrict__ b1,
    const float* __restrict__ W2, const float* __restrict__ b2,
    const float* __restrict__ W3, const float* __restrict__ b3,
    float* __restrict__ out) {
  __shared__ float feat[513];
  __shared__ float o1[256];
  __shared__ float o2[128];
  int t = threadIdx.x;
  for (int i = t; i < 512; i += 256) feat[i] = g[i];
  if (t == 0) feat[512] = y[1];
  __syncthreads();
  float a = b1[t];
  for (int k = 0; k < 513; k++) a += feat[k] * W1[k * 256 + t];
  o1[t] = fmaxf(a, 0.0f);
  __syncthreads();
  if (t < 128) {
    float a2 = b2[t];
    for (int k = 0; k < 256; k++) a2 += o1[k] * W2[k * 128 + t];
    o2[t] = fmaxf(a2, 0.0f);
  }
  __syncthreads();
  if (t == 0) {
    float a3 = b3[0];
    for (int k = 0; k < 128; k++) a3 += o2[k] * W3[k];
    out[0] = a3;
  }
}

// ============================================================================
extern "C" void kernel_launch(void* const* d_in, const int* in_sizes, int n_in,
                              void* d_out, int out_size, void* d_ws, size_t ws_size,
                              hipStream_t stream) {
  (void)in_sizes; (void)out_size; (void)ws_size;
  const float* X   = (const float*)d_in[0];
  const float* POS = (const float*)d_in[1];
  const float* Y   = (const float*)d_in[2];
  const float* L[42];
  const int* PERM;

  if (n_in >= 47) {               // pytree leaves flattened individually
    for (int i = 0; i < 42; i++) L[i] = (const float*)d_in[3 + i];
    PERM = (const int*)d_in[46];
  } else {                        // dict entries = concatenated blobs
    static const int sa1_sz[12] = {352,32,32,32, 1024,32,32,32, 2048,64,64,64};
    static const int sa2_sz[12] = {4288,64,64,64, 4096,64,64,64, 8192,128,128,128};
    static const int sa3_sz[12] = {16768,128,128,128, 32768,256,256,256, 131072,512,512,512};
    const float* base;
    size_t o;
    base = (const float*)d_in[3]; o = 0; for (int i = 0; i < 12; i++) { L[i]      = base + o; o += sa1_sz[i]; }
    base = (const float*)d_in[4]; o = 0; for (int i = 0; i < 12; i++) { L[12 + i] = base + o; o += sa2_sz[i]; }
    base = (const float*)d_in[5]; o = 0; for (int i = 0; i < 12; i++) { L[24 + i] = base + o; o += sa3_sz[i]; }
    base = (const float*)d_in[6]; L[36] = base; L[37] = base + 131328;
    base = (const float*)d_in[7]; L[38] = base; L[39] = base + 32768;
    base = (const float*)d_in[8]; L[40] = base; L[41] = base + 128;
    PERM = (const int*)d_in[10];
  }

  // ---- scratch layout (256B aligned) ----
  size_t off = 0;
  char* wsb = (char*)d_ws;
  auto alloc = [&](size_t bytes) -> char* {
    char* p = wsb + off;
    off += (bytes + 255) & ~(size_t)255;
    return p;
  };
  float* xp   = (float*)alloc(8192 * 8 * 4);
  float* posp = (float*)alloc(8192 * 3 * 4);
  float* q1   = (float*)alloc(4096 * 3 * 4);
  int*   nbr1 = (int*)  alloc(4096 * 64 * 4);
  int*   cnt1 = (int*)  alloc(256);
  float* x1   = (float*)alloc(2048 * 64 * 4);
  float* q2   = (float*)alloc(512 * 3 * 4);
  int*   nbr2 = (int*)  alloc(512 * 64 * 4);
  int*   cnt2 = (int*)  alloc(256);
  float* x2   = (float*)alloc(512 * 128 * 4);
  float* gbuf = (float*)alloc(512 * 4);
  float* gsum = (float*)alloc(512 * 4);
  float* gsq  = (float*)alloc(512 * 4);
  float* sA   = (float*)alloc(512 * 4);
  float* tA   = (float*)alloc(512 * 4);
  float* sB   = (float*)alloc(512 * 4);
  float* tB   = (float*)alloc(512 * 4);
  float* Wp   = (float*)alloc(256 * 512 * 4);
  float* bp   = (float*)alloc(512 * 4);
  float* bufA = (float*)alloc((size_t)262144 * 32 * 4);   // 33.5 MB
  float* bufB = (float*)alloc((size_t)262144 * 64 * 4);   // 67 MB

  // ================= SA1 =================
  permute_k<<<32, 256, 0, stream>>>(X, POS, PERM, xp, posp, 8192);
  fps_k<<<1, 1024, 0, stream>>>(posp, 8192, 4096, q1);
  zero_int_k<<<1, 1, 0, stream>>>(cnt1);
  ball_query_k<<<16, 256, 0, stream>>>(posp, 8192, q1, 4096, 0.04f, nbr1, cnt1);
  gather_sa1_k<<<1024, 256, 0, stream>>>(xp, posp, q1, nbr1, bufA);

  prep_wb_k<<<1, 32, 0, stream>>>(L[0], L[1], nullptr, nullptr, Wp, bp, 11, 12, 32, gsum, gsq);
  mlp_gemm_t<12, 32, 2><<<dim3(16384, 1), 32, 0, stream>>>(bufA, Wp, bp, bufB, nbr1, gsum, gsq);
  fin_stats_k<<<1, 32, 0, stream>>>(gsum, gsq, cnt1, 0, L[2], L[3], sA, tA);

  prep_wb_k<<<1, 32, 0, stream>>>(L[4], L[5], sA, tA, Wp, bp, 32, 32, 32, gsum, gsq);
  mlp_gemm_t<32, 32, 2><<<dim3(16384, 1), 32, 0, stream>>>(bufB, Wp, bp, bufA, nbr1, gsum, gsq);
  fin_stats_k<<<1, 32, 0, stream>>>(gsum, gsq, cnt1, 0, L[6], L[7], sB, tB);

  prep_wb_k<<<1, 64, 0, stream>>>(L[8], L[9], sB, tB, Wp, bp, 32, 32, 64, gsum, gsq);
  mlp_gemm_t<32, 64, 4><<<dim3(16384, 1), 32, 0, stream>>>(bufA, Wp, bp, bufB, nbr1, gsum, gsq);
  fin_stats_k<<<1, 64, 0, stream>>>(gsum, gsq, cnt1, 0, L[10], L[11], sA, tA);

  pc_max_k<<<2048, 64, 0, stream>>>(bufB, 64, nbr1, sA, tA, x1);   // x1[:2048]

  // ================= SA2 =================
  fps_k<<<1, 1024, 0, stream>>>(q1, 2048, 512, q2);                // pos1 = q1[:2048]
  zero_int_k<<<1, 1, 0, stream>>>(cnt2);
  ball_query_k<<<2, 256, 0, stream>>>(q1, 2048, q2, 512, 0.16f, nbr2, cnt2);
  gather_sa2_k<<<128, 256, 0, stream>>>(x1, q1, q2, nbr2, bufA);

  prep_wb_k<<<1, 64, 0, stream>>>(L[12], L[13], nullptr, nullptr, Wp, bp, 67, 68, 64, gsum, gsq);
  mlp_gemm_t<68, 64, 4><<<dim3(2048, 1), 32, 0, stream>>>(bufA, Wp, bp, bufB, nbr2, gsum, gsq);
  fin_stats_k<<<1, 64, 0, stream>>>(gsum, gsq, cnt2, 0, L[14], L[15], sA, tA);

  prep_wb_k<<<1, 64, 0, stream>>>(L[16], L[17], sA, tA, Wp, bp, 64, 64, 64, gsum, gsq);
  mlp_gemm_t<64, 64, 4><<<dim3(2048, 1), 32, 0, stream>>>(bufB, Wp, bp, bufA, nbr2, gsum, gsq);
  fin_stats_k<<<1, 64, 0, stream>>>(gsum, gsq, cnt2, 0, L[18], L[19], sB, tB);

  prep_wb_k<<<1, 128, 0, stream>>>(L[20], L[21], sB, tB, Wp, bp, 64, 64, 128, gsum, gsq);
  mlp_gemm_t<64, 128, 4><<<dim3(2048, 2), 32, 0, stream>>>(bufA, Wp, bp, bufB, nbr2, gsum, gsq);
  fin_stats_k<<<1, 128, 0, stream>>>(gsum, gsq, cnt2, 0, L[22], L[23], sA, tA);

  pc_max_k<<<512, 128, 0, stream>>>(bufB, 128, nbr2, sA, tA, x2);

  // ================= SA3 =================
  concat_sa3_k<<<264, 256, 0, stream>>>(x2, q2, bufA);

  prep_wb_k<<<1, 128, 0, stream>>>(L[24], L[25], nullptr, nullptr, Wp, bp, 131, 132, 128, gsum, gsq);
  mlp_gemm_t<132, 128, 4><<<dim3(32, 2), 32, 0, stream>>>(bufA, Wp, bp, bufB, nullptr, gsum, gsq);
  fin_stats_k<<<1, 128, 0, stream>>>(gsum, gsq, nullptr, 512, L[26], L[27], sA, tA);

  prep_wb_k<<<1, 256, 0, stream>>>(L[28], L[29], sA, tA, Wp, bp, 128, 128, 256, gsum, gsq);
  mlp_gemm_t<128, 256, 4><<<dim3(32, 4), 32, 0, stream>>>(bufB, Wp, bp, bufA, nullptr, gsum, gsq);
  fin_stats_k<<<1, 256, 0, stream>>>(gsum, gsq, nullptr, 512, L[30], L[31], sB, tB);

  prep_wb_k<<<1, 512, 0, stream>>>(L[32], L[33], sB, tB, Wp, bp, 256, 256, 512, gsum, gsq);
  mlp_gemm_t<256, 512, 4><<<dim3(32, 8), 32, 0, stream>>>(bufA, Wp, bp, bufB, nullptr, gsum, gsq);
  fin_stats_k<<<1, 512, 0, stream>>>(gsum, gsq, nullptr, 512, L[34], L[35], sA, tA);

  colmax_k<<<1, 512, 0, stream>>>(bufB, 512, 512, sA, tA, gbuf);

  // ================= Head =================
  head_k<<<1, 256, 0, stream>>>(gbuf, Y, L[36], L[37], L[38], L[39], L[40], L[41],
                                (float*)d_out);
}